// PNANet_order_28432683499893
// MI455X (gfx1250) — compile-verified
//
#include <hip/hip_runtime.h>
#include <math.h>
#include <float.h>

// ---------------------------------------------------------------------------
// PNA GNN forward for MI455X (gfx1250, wave32, WMMA).
// Edge pre-NN as fused f32-WMMA GEMM + atomic segment-{sum,sum2,min,max}.
// Min/max use an order-preserving u32 encoding -> branchless u32 atomics.
// ---------------------------------------------------------------------------

typedef float v2f __attribute__((ext_vector_type(2)));
typedef float v8f __attribute__((ext_vector_type(8)));

#define KT_MAX 176   // 2*Fin + 16 with Fin=80 (layer 0)

// Monotone float <-> u32 mapping: a < b (float) <=> enc(a) < enc(b) (unsigned)
static __device__ __forceinline__ unsigned encodeF(float v) {
    unsigned u = __float_as_uint(v);
    return ((int)u >= 0) ? (u | 0x80000000u) : ~u;
}
static __device__ __forceinline__ float decodeF(unsigned u) {
    unsigned b = ((int)u < 0) ? (u ^ 0x80000000u) : ~u;
    return __uint_as_float(b);
}
#define ENC_POS_FLTMAX 0xFF7FFFFFu   // encodeF(+FLT_MAX)  (init for min buffer)
#define ENC_NEG_FLTMAX 0x00800000u   // encodeF(-FLT_MAX)  (init for max buffer)

// ------------------------------ utility ------------------------------------
__global__ void fill_kernel(float* __restrict__ p, float v, int n) {
    int i = blockIdx.x * blockDim.x + threadIdx.x;
    if (i < n) p[i] = v;
}
__global__ void fill_u32_kernel(unsigned* __restrict__ p, unsigned v, int n) {
    int i = blockIdx.x * blockDim.x + threadIdx.x;
    if (i < n) p[i] = v;
}

__global__ void degree_kernel(const int* __restrict__ dst, float* __restrict__ deg, int E) {
    int i = blockIdx.x * blockDim.x + threadIdx.x;
    if (i < E) atomicAdd(&deg[dst[i]], 1.0f);
}

__global__ void logsum_kernel(const float* __restrict__ deg, float* __restrict__ out, int N) {
    __shared__ float red[256];
    int i = blockIdx.x * 256 + threadIdx.x;
    float v = (i < N) ? logf(deg[i] + 1.0f) : 0.0f;
    red[threadIdx.x] = v; __syncthreads();
    for (int s = 128; s > 0; s >>= 1) {
        if (threadIdx.x < s) red[threadIdx.x] += red[threadIdx.x + s];
        __syncthreads();
    }
    if (threadIdx.x == 0) atomicAdd(out, red[0]);
}

__global__ void node_scalars_kernel(const float* __restrict__ deg, const float* __restrict__ lsum,
                                    float* __restrict__ ampA, float* __restrict__ attA, int N) {
    int i = blockIdx.x * blockDim.x + threadIdx.x;
    if (i >= N) return;
    float avg = lsum[0] / (float)N;
    float ld  = logf(fmaxf(deg[i], 1.0f) + 1.0f);
    ampA[i] = ld / avg;
    attA[i] = avg / ld;
}

// h0[n, 0:64]  = x @ nodeW + nodeB ; h0[n, 64:80] = relu(extra @ extraW + extraB)
__global__ void embed_h0_kernel(const float* __restrict__ x, const float* __restrict__ extra,
                                const float* __restrict__ nW, const float* __restrict__ nB,
                                const float* __restrict__ xW, const float* __restrict__ xB,
                                float* __restrict__ h0, int N) {
    int i = blockIdx.x * blockDim.x + threadIdx.x;
    if (i >= N * 80) return;
    int n = i / 80, c = i - n * 80;
    float a;
    if (c < 64) {
        a = nB[c];
        #pragma unroll
        for (int k = 0; k < 32; ++k) a += x[n * 32 + k] * nW[k * 64 + c];
    } else {
        int cc = c - 64;
        a = xB[cc];
        #pragma unroll
        for (int k = 0; k < 16; ++k) a += extra[n * 16 + k] * xW[k * 16 + cc];
        a = fmaxf(a, 0.0f);
    }
    h0[i] = a;
}

__global__ void embed_e_kernel(const float* __restrict__ ea, const float* __restrict__ eW,
                               const float* __restrict__ eB, float* __restrict__ eEmb, int E) {
    int i = blockIdx.x * blockDim.x + threadIdx.x;
    if (i >= E * 16) return;
    int e = i >> 4, c = i & 15;
    float a = eB[c];
    #pragma unroll
    for (int k = 0; k < 8; ++k) a += ea[e * 8 + k] * eW[k * 16 + c];
    eEmb[i] = a;
}

// Fold edge_enc into pre_W (see previous round comments).
__global__ void build_wcat_kernel(const float* __restrict__ preW, const float* __restrict__ preB,
                                  const float* __restrict__ encW, const float* __restrict__ encB,
                                  float* __restrict__ Wcat, float* __restrict__ bcat, int Fin) {
    int NCOL = 4 * Fin, KT = 2 * Fin + 16;
    int idx = blockIdx.x * blockDim.x + threadIdx.x;
    if (idx >= KT * NCOL) return;
    int r = idx / NCOL, col = idx - r * NCOL;
    int t = col / Fin, o = col - t * Fin;
    const float* pwt = preW + (size_t)t * 3 * Fin * Fin;
    float v;
    if (r < 2 * Fin) {
        v = pwt[r * Fin + o];
    } else {
        int j = r - 2 * Fin;
        v = 0.0f;
        for (int k = 0; k < Fin; ++k) v += encW[j * Fin + k] * pwt[(2 * Fin + k) * Fin + o];
    }
    Wcat[r * NCOL + col] = v;
    if (r == 0) {
        float b = preB[t * Fin + o];
        for (int k = 0; k < Fin; ++k) b += encB[k] * pwt[(2 * Fin + k) * Fin + o];
        bcat[col] = b;
    }
}

// ------------------- fused edge GEMM (WMMA) + aggregation ------------------
// Block = 128 threads (4 waves). blockIdx.y selects a 64-column group; each
// wave owns a 16-col subtile. Compile-time Fin -> fully unrolled K-loop of
// V_WMMA_F32_16X16X4_F32 so LDS loads pipeline ahead of the WMMA chain.
template <int Fin>
__global__ __launch_bounds__(128)
void edge_gemm_kernel(const float* __restrict__ h, int hstr,
                      const float* __restrict__ eEmb,
                      const int* __restrict__ srcI, const int* __restrict__ dstI,
                      const float* __restrict__ Wcat, const float* __restrict__ bcat,
                      float* __restrict__ Ssum, float* __restrict__ Ssq,
                      unsigned* __restrict__ SminK, unsigned* __restrict__ SmaxK,
                      int Etiles) {
    constexpr int KT   = 2 * Fin + 16;
    constexpr int NCOL = 4 * Fin;

    __shared__ float Wlds[KT * 64];          // <= 45 KB
    __shared__ float featLds[16 * (KT + 1)]; // <= 11.3 KB, padded stride
    __shared__ int   sDst[16];
    __shared__ int   sSrc[16];

    const int tid  = threadIdx.x;
    const int lane = tid & 31;
    const int wave = tid >> 5;
    const int c0   = blockIdx.y * 64;        // column-group base
    const int cbase = c0 + (wave << 4);      // this wave's 16-col base

    // stage weight slice [KT x 64] in LDS (reused across all edge tiles)
    for (int idx = tid; idx < KT * 64; idx += 128) {
        int r = idx >> 6, c = idx & 63;
        Wlds[idx] = Wcat[r * NCOL + c0 + c];
    }

    const float bv    = bcat[cbase + (lane & 15)];
    const int   mrow  = lane & 15;            // A row / B col within tile
    const int   khalf = (lane >> 4) << 1;     // 0 or 2 (f32 WMMA lane split)

    for (int et = blockIdx.x; et < Etiles; et += gridDim.x) {
        __syncthreads();
        if (tid < 16) { sDst[tid] = dstI[et * 16 + tid]; sSrc[tid] = srcI[et * 16 + tid]; }
        __syncthreads();

        // gather feat[m][k] = [ h[dst] (Fin) | h[src] (Fin) | e_emb (16) ]
        #pragma unroll
        for (int m = 0; m < 16; ++m) {
            int ed = sDst[m], es = sSrc[m];
            for (int k = tid; k < KT; k += 128) {
                float v;
                if (k < Fin)            v = h[ed * hstr + k];
                else if (k < 2 * Fin)   v = h[es * hstr + (k - Fin)];
                else                    v = eEmb[(et * 16 + m) * 16 + (k - 2 * Fin)];
                featLds[m * (KT + 1) + k] = v;
            }
        }
        __syncthreads();

        v8f acc;
        #pragma unroll
        for (int j = 0; j < 8; ++j) acc[j] = bv;

        #pragma unroll
        for (int k0 = 0; k0 < KT; k0 += 4) {
            const int ka = k0 + khalf;
            v2f a, b;
            a.x = featLds[mrow * (KT + 1) + ka];
            a.y = featLds[mrow * (KT + 1) + ka + 1];
            b.x = Wlds[ka * 64 + (wave << 4) + mrow];
            b.y = Wlds[(ka + 1) * 64 + (wave << 4) + mrow];
            acc = __builtin_amdgcn_wmma_f32_16x16x4_f32(
                false, a, false, b, (short)0, acc, false, false);
        }

        // scatter: C layout -> VGPR j holds rows M=j (lanes 0-15), M=j+8 (16-31)
        const int colg = cbase + (lane & 15);
        #pragma unroll
        for (int j = 0; j < 8; ++j) {
            int m = j + ((lane >> 4) << 3);
            int d = sDst[m];
            float v = acc[j];
            size_t off = (size_t)d * NCOL + colg;
            atomicAdd(&Ssum[off], v);
            atomicAdd(&Ssq[off],  v * v);
            unsigned key = encodeF(v);
            atomicMax(&SmaxK[off], key);   // branchless global_atomic_max_u32
            atomicMin(&SminK[off], key);   // branchless global_atomic_min_u32
        }
    }
}

// --------------------- node post-NN + lin + BN partials --------------------
// 256 threads = 4 nodes x 64 (tower,out) channels.
template <int Fin>
__global__ __launch_bounds__(256)
void post_kernel(const float* __restrict__ h, int hstr,
                 const float* __restrict__ Ssum, const float* __restrict__ Ssq,
                 const unsigned* __restrict__ SminK, const unsigned* __restrict__ SmaxK,
                 const float* __restrict__ deg, const float* __restrict__ ampA,
                 const float* __restrict__ attA,
                 const float* __restrict__ postW, const float* __restrict__ postB,
                 const float* __restrict__ linW, const float* __restrict__ linB,
                 float* __restrict__ y, float* __restrict__ bn_sum, float* __restrict__ bn_sq,
                 int N) {
    __shared__ float sh_out[4 * 64];
    __shared__ float sh_y[4 * 64];
    const int tid = threadIdx.x;
    const int nl = tid >> 6, ch = tid & 63;
    const int t = ch >> 4, o = ch & 15;
    const int n = blockIdx.x * 4 + nl;
    constexpr int NCOL = 4 * Fin;
    constexpr int R = Fin * 16;  // element stride between 13F blocks of post_W

    float dg = deg[n];
    float inv_cnt = 1.0f / fmaxf(dg, 1.0f);
    float amp = ampA[n], att = attA[n];
    bool  has = dg > 0.0f;

    const float* pw = postW + (size_t)t * 13 * Fin * 16 + o;
    const size_t sb = (size_t)n * NCOL + t * Fin;
    float acc = postB[t * 16 + o];

    for (int f = 0; f < Fin; ++f) {
        float s = Ssum[sb + f], q = Ssq[sb + f];
        float mean = s * inv_cnt;
        float var  = q * inv_cnt - mean * mean;
        float stdv = sqrtf(fmaxf(var, 0.0f) + 1e-5f);
        float mn = has ? decodeF(SminK[sb + f]) : 0.0f;
        float mx = has ? decodeF(SmaxK[sb + f]) : 0.0f;
        const float* pwf = pw + f * 16;
        acc += h[n * hstr + f] * pwf[0]
             + mean * pwf[1 * R] + mn * pwf[2 * R] + mx * pwf[3 * R] + stdv * pwf[4 * R]
             + amp * (mean * pwf[5 * R] + mn * pwf[6 * R] + mx * pwf[7 * R] + stdv * pwf[8 * R])
             + att * (mean * pwf[9 * R] + mn * pwf[10 * R] + mx * pwf[11 * R] + stdv * pwf[12 * R]);
    }
    sh_out[nl * 64 + ch] = acc;
    __syncthreads();

    float yv = linB[ch];
    #pragma unroll 8
    for (int k = 0; k < 64; ++k) yv += sh_out[nl * 64 + k] * linW[k * 64 + ch];
    y[(size_t)n * 64 + ch] = yv;
    sh_y[nl * 64 + ch] = yv;
    __syncthreads();

    if (nl == 0) {
        float s = 0.0f, sq = 0.0f;
        #pragma unroll
        for (int i = 0; i < 4; ++i) { float v = sh_y[i * 64 + ch]; s += v; sq += v * v; }
        atomicAdd(&bn_sum[ch], s);
        atomicAdd(&bn_sq[ch],  sq);
    }
}

__global__ void bn_relu_kernel(const float* __restrict__ y,
                               const float* __restrict__ bn_sum, const float* __restrict__ bn_sq,
                               const float* __restrict__ gamma, const float* __restrict__ beta,
                               float* __restrict__ hout, int N) {
    int i = blockIdx.x * blockDim.x + threadIdx.x;
    if (i >= N * 64) return;
    int c = i & 63;
    float invN = 1.0f / (float)N;
    float mu = bn_sum[c] * invN;
    float var = bn_sq[c] * invN - mu * mu;
    float v = (y[i] - mu) * rsqrtf(var + 1e-5f) * gamma[c] + beta[c];
    hout[i] = fmaxf(v, 0.0f);
}

__global__ void pool_kernel(const float* __restrict__ h, const float* __restrict__ mask,
                            const int* __restrict__ batch, float* __restrict__ g, int N) {
    int i = blockIdx.x * blockDim.x + threadIdx.x;
    if (i >= N * 64) return;
    int n = i >> 6, c = i & 63;
    atomicAdd(&g[batch[n] * 64 + c], h[i] * mask[n]);
}

__global__ void mlp_kernel(const float* __restrict__ g,
                           const float* __restrict__ W1, const float* __restrict__ b1,
                           const float* __restrict__ W2, const float* __restrict__ b2,
                           const float* __restrict__ W3, const float* __restrict__ b3,
                           float* __restrict__ out) {
    __shared__ float s0[64], s1[64], s2[32];
    int gi = blockIdx.x, c = threadIdx.x;
    s0[c] = g[gi * 64 + c];
    __syncthreads();
    float a = b1[c];
    #pragma unroll 8
    for (int k = 0; k < 64; ++k) a += s0[k] * W1[k * 64 + c];
    s1[c] = fmaxf(a, 0.0f);
    __syncthreads();
    if (c < 32) {
        float a2 = b2[c];
        #pragma unroll 8
        for (int k = 0; k < 64; ++k) a2 += s1[k] * W2[k * 32 + c];
        s2[c] = fmaxf(a2, 0.0f);
    }
    __syncthreads();
    if (c == 0) {
        float a3 = b3[0];
        #pragma unroll
        for (int k = 0; k < 32; ++k) a3 += s2[k] * W3[k];
        out[gi] = a3;
    }
}

// ------------------------------- launcher ----------------------------------
static inline int cdiv(int a, int b) { return (a + b - 1) / b; }

extern "C" void kernel_launch(void* const* d_in, const int* in_sizes, int n_in,
                              void* d_out, int out_size, void* d_ws, size_t ws_size,
                              hipStream_t stream) {
    const int N = in_sizes[0] / 32;
    const int E = in_sizes[2] / 8;
    const int G = out_size;   // 64 graphs x 1 class

    // Inputs in setup_inputs() insertion order (depth-first; (W,b) tuples flat):
    const float* x     = (const float*)d_in[0];
    const float* extra = (const float*)d_in[1];
    const float* eattr = (const float*)d_in[2];
    const float* mask  = (const float*)d_in[3];
    const float* nW = (const float*)d_in[4];
    const float* nB = (const float*)d_in[5];
    const float* xW = (const float*)d_in[6];
    const float* xB = (const float*)d_in[7];
    const float* eW = (const float*)d_in[8];
    const float* eB = (const float*)d_in[9];
    // convs: base 10 + 10*l : encW, encB, preW, preB, postW, postB, linW, linB, gamma, beta
    const float* m0W = (const float*)d_in[40];
    const float* m0B = (const float*)d_in[41];
    const float* m1W = (const float*)d_in[42];
    const float* m1B = (const float*)d_in[43];
    const float* m2W = (const float*)d_in[44];
    const float* m2B = (const float*)d_in[45];
    const int* eidx = (const int*)d_in[46];
    const int* bids = (const int*)d_in[47];
    const int* srcI = eidx;          // edge_index[0]
    const int* dstI = eidx + E;      // edge_index[1]

    float* ws = (float*)d_ws;
    size_t off = 0;
    auto wsalloc = [&](size_t nElem) { float* p = ws + off; off += nElem; return p; };
    float* h0    = wsalloc((size_t)N * 80);
    float* hA    = wsalloc((size_t)N * 64);
    float* hB    = wsalloc((size_t)N * 64);
    float* eEmb  = wsalloc((size_t)E * 16);
    float* Ssum  = wsalloc((size_t)N * 320);
    float* Ssq   = wsalloc((size_t)N * 320);
    unsigned* SminK = (unsigned*)wsalloc((size_t)N * 320);
    unsigned* SmaxK = (unsigned*)wsalloc((size_t)N * 320);
    float* Wcat  = wsalloc((size_t)KT_MAX * 320);
    float* bcat  = wsalloc(320);
    float* ybuf  = wsalloc((size_t)N * 64);
    float* deg   = wsalloc(N);
    float* ampA  = wsalloc(N);
    float* attA  = wsalloc(N);
    float* lsum  = wsalloc(1);
    float* bnS   = wsalloc(64);
    float* bnQ   = wsalloc(64);
    float* gbuf  = wsalloc((size_t)G * 64);

    const int B = 256;

    // degree scalers
    fill_kernel<<<cdiv(N, B), B, 0, stream>>>(deg, 0.0f, N);
    fill_kernel<<<1, 32, 0, stream>>>(lsum, 0.0f, 1);
    degree_kernel<<<cdiv(E, B), B, 0, stream>>>(dstI, deg, E);
    logsum_kernel<<<cdiv(N, 256), 256, 0, stream>>>(deg, lsum, N);
    node_scalars_kernel<<<cdiv(N, B), B, 0, stream>>>(deg, lsum, ampA, attA, N);

    // embeddings
    embed_h0_kernel<<<cdiv(N * 80, B), B, 0, stream>>>(x, extra, nW, nB, xW, xB, h0, N);
    embed_e_kernel<<<cdiv(E * 16, B), B, 0, stream>>>(eattr, eW, eB, eEmb, E);

    const float* hin = h0;
    int hstr = 80;
    for (int l = 0; l < 3; ++l) {
        const int Fin = (l == 0) ? 80 : 64;
        const int KT = 2 * Fin + 16, NCOL = 4 * Fin;
        const int bi = 10 + 10 * l;
        const float* encW  = (const float*)d_in[bi + 0];
        const float* encB  = (const float*)d_in[bi + 1];
        const float* preW  = (const float*)d_in[bi + 2];
        const float* preB  = (const float*)d_in[bi + 3];
        const float* postW = (const float*)d_in[bi + 4];
        const float* postB = (const float*)d_in[bi + 5];
        const float* linW  = (const float*)d_in[bi + 6];
        const float* linB  = (const float*)d_in[bi + 7];
        const float* gamma = (const float*)d_in[bi + 8];
        const float* beta  = (const float*)d_in[bi + 9];
        float* hout = (l == 0) ? hA : ((l == 1) ? hB : hA);

        build_wcat_kernel<<<cdiv(KT * NCOL, B), B, 0, stream>>>(preW, preB, encW, encB, Wcat, bcat, Fin);
        fill_kernel<<<cdiv(N * NCOL, B), B, 0, stream>>>(Ssum, 0.0f, N * NCOL);
        fill_kernel<<<cdiv(N * NCOL, B), B, 0, stream>>>(Ssq, 0.0f, N * NCOL);
        fill_u32_kernel<<<cdiv(N * NCOL, B), B, 0, stream>>>(SminK, ENC_POS_FLTMAX, N * NCOL);
        fill_u32_kernel<<<cdiv(N * NCOL, B), B, 0, stream>>>(SmaxK, ENC_NEG_FLTMAX, N * NCOL);
        fill_kernel<<<1, 64, 0, stream>>>(bnS, 0.0f, 64);
        fill_kernel<<<1, 64, 0, stream>>>(bnQ, 0.0f, 64);

        dim3 egrid(512, NCOL / 64);
        if (Fin == 80) {
            edge_gemm_kernel<80><<<egrid, 128, 0, stream>>>(hin, hstr, eEmb, srcI, dstI,
                                                            Wcat, bcat, Ssum, Ssq, SminK, SmaxK, E / 16);
            post_kernel<80><<<N / 4, 256, 0, stream>>>(hin, hstr, Ssum, Ssq, SminK, SmaxK,
                                                       deg, ampA, attA, postW, postB, linW, linB,
                                                       ybuf, bnS, bnQ, N);
        } else {
            edge_gemm_kernel<64><<<egrid, 128, 0, stream>>>(hin, hstr, eEmb, srcI, dstI,
                                                            Wcat, bcat, Ssum, Ssq, SminK, SmaxK, E / 16);
            post_kernel<64><<<N / 4, 256, 0, stream>>>(hin, hstr, Ssum, Ssq, SminK, SmaxK,
                                                       deg, ampA, attA, postW, postB, linW, linB,
                                                       ybuf, bnS, bnQ, N);
        }
        bn_relu_kernel<<<cdiv(N * 64, B), B, 0, stream>>>(ybuf, bnS, bnQ, gamma, beta, hout, N);

        hin = hout;
        hstr = 64;
    }

    fill_kernel<<<cdiv(G * 64, B), B, 0, stream>>>(gbuf, 0.0f, G * 64);
    pool_kernel<<<cdiv(N * 64, B), B, 0, stream>>>(hin, mask, bids, gbuf, N);
    mlp_kernel<<<G, 64, 0, stream>>>(gbuf, m0W, m0B, m1W, m1B, m2W, m2B, (float*)d_out);
}